// LocalPooling1D_80848464380246
// MI455X (gfx1250) — compile-verified
//
#include <hip/hip_runtime.h>
#include <stdint.h>

typedef __attribute__((ext_vector_type(2))) float v2f;
typedef __attribute__((ext_vector_type(8))) float v8f;
typedef __attribute__((ext_vector_type(4))) int   v4i;

#define B_   32
#define T_   8192
#define C_   256
#define P_   65
#define NSEG 64
#define CHUNK 48                          // tokens per LDS tile
#define LDS_STRIDE 136                    // 128 data + 8 pad floats -> conflict-free
#define ISSUES ((CHUNK * 128) / (256 * 4))  // 6 B128 issues per thread per chunk

#define GLOBAL_AS __attribute__((address_space(1)))
#define LDS_AS    __attribute__((address_space(3)))

#if defined(__has_builtin)
#if __has_builtin(__builtin_amdgcn_global_load_async_to_lds_b128)
#define HAVE_ASYNC_LDS 1
#endif
#endif

#ifdef HAVE_ASYNC_LDS
#define S_WAIT_ASYNCCNT(N) asm volatile("s_wait_asynccnt %0" :: "n"(N) : "memory")
#endif

// Issue one chunk of token rows [t0, t0+CHUNK) x 128 channels into `buf`.
// Rows past the valid range load from a clamped valid address (finite garbage);
// the WMMA membership weight is exactly 0 for those tokens, so they contribute 0.
__device__ __forceinline__ void issue_chunk(const float* __restrict__ x, int b, int t0,
                                            int cbase, int tid, float* buf)
{
#pragma unroll
    for (int s = 0; s < ISSUES; ++s) {
        const int id  = tid + s * 256;   // float4 slot, 32 per token row
        const int tok = id >> 5;
        const int cq  = (id & 31) << 2;
        int tg = t0 + tok;
        tg = tg < T_ ? tg : T_ - 1;      // clamp: always a valid 16B-aligned address
        const float* gp = x + ((size_t)(b * T_ + tg) * C_ + cbase + cq);
        float* lp = buf + tok * LDS_STRIDE + cq;
#ifdef HAVE_ASYNC_LDS
        __builtin_amdgcn_global_load_async_to_lds_b128(
            (GLOBAL_AS v4i*)gp, (LDS_AS v4i*)lp, /*offset=*/0, /*cpol=*/0);
#else
        *(float4*)lp = *(const float4*)gp;   // sync fallback: reg round-trip
#endif
    }
}

// WMMA accumulate over one full chunk: D(16seg x 16ch) += A(16x4 membership) * B(4x16).
__device__ __forceinline__ void compute_chunk(const float* buf, int t0, int sL, int sR,
                                              int cw, int n, int khalf, v8f& acc)
{
    for (int k = 0; k < CHUNK / 4; ++k) {
        const int lt = (k << 2) + (khalf << 1);   // local token of this lane's first K
        const int gt = t0 + lt;                   // global token index
        v2f a, bm;
        a.x = (gt     >= sL && gt     < sR) ? 1.0f : 0.0f;
        a.y = (gt + 1 >= sL && gt + 1 < sR) ? 1.0f : 0.0f;
        bm.x = buf[lt       * LDS_STRIDE + cw + n];
        bm.y = buf[(lt + 1) * LDS_STRIDE + cw + n];
        acc = __builtin_amdgcn_wmma_f32_16x16x4_f32(
                  false, a, false, bm, (short)0, acc, false, false);
    }
}

__global__ __launch_bounds__(256) void segmean_wmma_kernel(
    const float* __restrict__ x, const int* __restrict__ pos, float* __restrict__ y)
{
    const int tid   = threadIdx.x;
    const int lane  = tid & 31;          // wave32
    const int wave  = tid >> 5;          // 8 waves
    const int bid   = blockIdx.x;        // 0..255
    const int b      = bid >> 3;
    const int segblk = (bid >> 1) & 3;
    const int chalf  = bid & 1;
    const int j0     = segblk * 16;
    const int cbase  = chalf * 128;      // first global channel of this workgroup
    const int cw     = wave * 16;        // channel offset inside LDS tile for this wave
    const int n      = lane & 15;        // N index within 16-channel block
    const int khalf  = lane >> 4;        // 0: K=0,1  1: K=2,3

    __shared__ float tile[2][CHUNK * LDS_STRIDE];   // 2 x 26112 B

    const int* posb = pos + b * P_;
    const int m_a   = lane & 15;
    const int sL    = posb[j0 + m_a];
    const int sR    = posb[j0 + m_a + 1];
    const int tstart = posb[j0];
    const int tend   = posb[j0 + 16];

    v8f acc = {0.f, 0.f, 0.f, 0.f, 0.f, 0.f, 0.f, 0.f};

    const int nch = (tend - tstart + CHUNK - 1) / CHUNK;   // uniform per workgroup

#ifdef HAVE_ASYNC_LDS
    if (nch > 0)
        issue_chunk(x, b, tstart, cbase, tid, tile[0]);    // preload chunk 0
    for (int i = 0; i < nch; ++i) {
        float* cur = tile[i & 1];
        if (i + 1 < nch) {
            // overlap: next chunk streams into the other buffer while we compute
            issue_chunk(x, b, tstart + (i + 1) * CHUNK, cbase, tid, tile[(i + 1) & 1]);
            S_WAIT_ASYNCCNT(ISSUES);     // chunk i complete; chunk i+1 still in flight
        } else {
            S_WAIT_ASYNCCNT(0);
        }
        __syncthreads();                 // all waves' chunk-i data landed in LDS
        compute_chunk(cur, tstart + i * CHUNK, sL, sR, cw, n, khalf, acc);
        __syncthreads();                 // done reading `cur` before it is refilled
    }
#else
    for (int i = 0; i < nch; ++i) {
        __syncthreads();
        issue_chunk(x, b, tstart + i * CHUNK, cbase, tid, tile[0]);
        __syncthreads();
        compute_chunk(tile[0], tstart + i * CHUNK, sL, sR, cw, n, khalf, acc);
    }
#endif

    // D layout: lanes 0-15 hold rows M=v, lanes 16-31 rows M=v+8; N = lane&15.
#pragma unroll
    for (int v = 0; v < 8; ++v) {
        const int m   = v + (khalf << 3);
        const int cnt = posb[j0 + m + 1] - posb[j0 + m];
        const float inv = 1.0f / (float)max(cnt, 1);   // empty segment -> 0
        const size_t oi = (size_t)(b * NSEG + j0 + m) * C_ + cbase + cw + n;
        y[oi] = acc[v] * inv;
    }
}

extern "C" void kernel_launch(void* const* d_in, const int* in_sizes, int n_in,
                              void* d_out, int out_size, void* d_ws, size_t ws_size,
                              hipStream_t stream) {
    const float* x  = (const float*)d_in[0];
    const int* pos  = (const int*)d_in[1];
    float* y        = (float*)d_out;
    // 32 batches * 4 segment-blocks * 2 channel-halves = 256 workgroups
    segmean_wmma_kernel<<<dim3(B_ * 4 * 2), dim3(256), 0, stream>>>(x, pos, y);
}